// SWTLoss_61770219651222
// MI455X (gfx1250) — compile-verified
//
#include <hip/hip_runtime.h>

typedef float v2f __attribute__((ext_vector_type(2)));
typedef float v8f __attribute__((ext_vector_type(8)));

#define TILE 32
#define DT   35   // d tile edge (32 + 3 halo)
#define AT   34   // aa0 tile edge (32 + 2 halo)
#define LDW  36   // padded LDS row width (bank-conflict padding)

// Wave32 cross-lane sum using V_WMMA_F32_16X16X4_F32.
// A-layout (ISA 7.12.2): VGPR0 lanes0-15 -> A(m,0)=acc[m]; lanes16-31 -> A(m,2)=acc[m+16].
// With B = all-ones and VGPR1 = 0:  D(m,n) = acc[m] + acc[m+16] for every n.
// Summing the 8 D VGPRs per lane leaves sum(acc[0..7]) + sum(acc[16..23]) in lanes 0-15 and
// sum(acc[8..15]) + sum(acc[24..31]) in lanes 16-31 (each uniform within its half).
__device__ __forceinline__ float wmma_halfwave_sum(float acc) {
    v2f a; a.x = acc;  a.y = 0.0f;
    v2f b; b.x = 1.0f; b.y = 1.0f;
    v8f cz = {};
    v8f d = __builtin_amdgcn_wmma_f32_16x16x4_f32(
        /*neg_a=*/false, a, /*neg_b=*/false, b,
        /*c_mod=*/(short)0, cz, /*reuse_a=*/false, /*reuse_b=*/false);
    return ((d[0] + d[1]) + (d[2] + d[3])) + ((d[4] + d[5]) + (d[6] + d[7]));
}

__global__ __launch_bounds__(256)
void swt_partial(const float* __restrict__ pred, const float* __restrict__ targ,
                 float* __restrict__ part) {
    __shared__ float sd[DT][LDW];
    __shared__ float sa[AT][LDW];
    __shared__ float sred[16];

    const int   tid   = threadIdx.x;
    const float C     = 0.70710678118654752440f;  // np.float32(2**-0.5)
    const size_t plane = (size_t)blockIdx.z * (512u * 512u);
    const int   row0  = blockIdx.y * TILE;
    const int   col0  = blockIdx.x * TILE;

    // ---- Phase 1: d = pred - target tile (wrapped) into LDS ----
    for (int idx = tid; idx < DT * DT; idx += 256) {
        int r = idx / DT;
        int c = idx - r * DT;
        int gr = (row0 + r) & 511;
        int gc = (col0 + c) & 511;
        size_t g = plane + (size_t)gr * 512 + (size_t)gc;
        sd[r][c] = pred[g] - targ[g];
    }
    __syncthreads();

    float acc = 0.0f;

    // ---- Phase 2: aa0 tile (34x34) + level-0 bands on inner 32x32 ----
    for (int idx = tid; idx < AT * AT; idx += 256) {
        int r = idx / AT;
        int c = idx - r * AT;
        float d00 = sd[r][c];
        float d10 = sd[r + 1][c];
        float d01 = sd[r][c + 1];
        float d11 = sd[r + 1][c + 1];
        float lo0 = (d00 + d10) * C;   // row (H) lowpass at (r,c)
        float lo1 = (d01 + d11) * C;   // row lowpass at (r,c+1)
        float aa  = (lo0 + lo1) * C;
        sa[r][c] = aa;
        if (r < TILE && c < TILE) {
            float hi0 = (d00 - d10) * C;
            float hi1 = (d01 - d11) * C;
            float hh  = (lo0 - lo1) * C;
            float vv  = (hi0 + hi1) * C;
            float dd  = (hi0 - hi1) * C;
            acc += 0.5f * fabsf(aa) + 1.5f * fabsf(hh)
                 + 1.5f * fabsf(vv) + 2.0f * fabsf(dd);
        }
    }
    __syncthreads();

    // ---- Phase 3: level-1 bands (dilation s=2) from aa0 tile ----
#pragma unroll
    for (int k = 0; k < 4; ++k) {
        int idx = tid + k * 256;            // 0..1023 over inner 32x32
        int r = idx >> 5;
        int c = idx & 31;
        float A00 = sa[r][c];
        float A20 = sa[r + 2][c];
        float A02 = sa[r][c + 2];
        float A22 = sa[r + 2][c + 2];
        float lo0 = (A00 + A20) * C;
        float lo1 = (A02 + A22) * C;
        float hi0 = (A00 - A20) * C;
        float hi1 = (A02 - A22) * C;
        float aa  = (lo0 + lo1) * C;
        float hh  = (lo0 - lo1) * C;
        float vv  = (hi0 + hi1) * C;
        float dd  = (hi0 - hi1) * C;
        acc += 0.5f * fabsf(aa) + 1.5f * fabsf(hh)
             + 1.5f * fabsf(vv) + 2.0f * fabsf(dd);
    }

    // ---- Phase 4: WMMA wave reduction, deterministic block combine ----
    float s = wmma_halfwave_sum(acc);
    int lane = tid & 31;
    int wave = tid >> 5;
    if (lane == 0)  sred[2 * wave + 0] = s;
    if (lane == 16) sred[2 * wave + 1] = s;
    __syncthreads();
    if (tid == 0) {
        float t = 0.0f;
#pragma unroll
        for (int i = 0; i < 16; ++i) t += sred[i];
        part[(size_t)blockIdx.z * 256 + blockIdx.y * 16 + blockIdx.x] = t;
    }
}

__global__ __launch_bounds__(256)
void swt_final(const float* __restrict__ part, int n, float* __restrict__ out) {
    __shared__ float sred[16];
    int tid = threadIdx.x;
    float acc = 0.0f;
    for (int i = tid; i < n; i += 256) acc += part[i];  // fixed order: deterministic
    float s = wmma_halfwave_sum(acc);
    int lane = tid & 31;
    int wave = tid >> 5;
    if (lane == 0)  sred[2 * wave + 0] = s;
    if (lane == 16) sred[2 * wave + 1] = s;
    __syncthreads();
    if (tid == 0) {
        float t = 0.0f;
#pragma unroll
        for (int i = 0; i < 16; ++i) t += sred[i];
        // total / (N * level * 4), N = 16*3*512*512 = 12582912
        out[0] = t * (1.0f / (12582912.0f * 8.0f));
    }
}

extern "C" void kernel_launch(void* const* d_in, const int* in_sizes, int n_in,
                              void* d_out, int out_size, void* d_ws, size_t ws_size,
                              hipStream_t stream) {
    const float* pred = (const float*)d_in[0];
    const float* targ = (const float*)d_in[1];
    float* out  = (float*)d_out;
    float* part = (float*)d_ws;   // 12288 floats = 48 KB of scratch

    dim3 grid(16, 16, 48);        // 32x32 tiles over 512x512, 48 planes
    swt_partial<<<grid, 256, 0, stream>>>(pred, targ, part);
    swt_final<<<1, 256, 0, stream>>>(part, 16 * 16 * 48, out);
}